// MLPAttention_71648644432025
// MI455X (gfx1250) — compile-verified
//
#include <hip/hip_runtime.h>
#include <hip/hip_bf16.h>
#include <math.h>

typedef __attribute__((ext_vector_type(2))) float v2f;
typedef __attribute__((ext_vector_type(8))) float v8f;

// ---------------------------------------------------------------------------
// f32 WMMA 16x16x4: A is 16x4 (2 VGPRs/lane), B is 4x16 (2 VGPRs), C/D 8 VGPRs.
// Guarded so a missing/mismatched builtin still compiles (histogram tells us).
// ---------------------------------------------------------------------------
static __device__ inline v8f wmma4(v2f a, v2f b, v8f c) {
#if defined(__has_builtin) && __has_builtin(__builtin_amdgcn_wmma_f32_16x16x4_f32)
  return __builtin_amdgcn_wmma_f32_16x16x4_f32(false, a, false, b, (short)0, c,
                                               false, false);
#else
  return c;  // fallback: compile-only sentinel (wmma=0 in histogram)
#endif
}

// ---------------------------------------------------------------------------
// K1: qp[q][h] = sum_d W1[h,d]   * query[d,q]          (tiles 0..255)
//     kpb[k][h] = sum_d W1[h,64+d]* key[d,k] + b1[h]   (tiles 256..767)
// One wave per 16x16 output tile, K-loop of 4 over d=64.
// A layout (32-bit 16x4): lanes 0-15: M=lane, v0=K0,v1=K1; lanes16-31: K2,K3.
// B layout (4x16): v0 = rows K0(l0-15)/K2(l16-31), v1 = rows K1/K3.
// C layout: vgpr r, lane L -> M = r + 8*(L>=16), N = L%16.
// ---------------------------------------------------------------------------
__global__ void proj_kernel(const float* __restrict__ query,
                            const float* __restrict__ key,
                            const float* __restrict__ W1,
                            const float* __restrict__ b1,
                            float* __restrict__ qp,
                            float* __restrict__ kpb) {
  int tile = blockIdx.x;
  const int lane = threadIdx.x;
  const int lm = lane & 15;
  const int hi = lane >> 4;

  const float* src;
  float* dst;
  int ncols, woff;
  bool isQ = (tile < 256);
  if (isQ) { src = query; dst = qp;  ncols = 1024; woff = 0;  }
  else     { tile -= 256; src = key; dst = kpb; ncols = 2048; woff = 64; }

  const int r0 = (tile >> 2) * 16;   // q/k tile origin
  const int h0 = (tile & 3) * 16;    // h tile origin

  v8f c = {};
  for (int d0 = 0; d0 < 64; d0 += 4) {
    const int ka = d0 + hi * 2;
    v2f a, b;
    a.x = src[(ka + 0) * ncols + r0 + lm];
    a.y = src[(ka + 1) * ncols + r0 + lm];
    b.x = W1[(h0 + lm) * 128 + woff + ka + 0];
    b.y = W1[(h0 + lm) * 128 + woff + ka + 1];
    c = wmma4(a, b, c);
  }
  const float bias = isQ ? 0.0f : b1[h0 + lm];
#pragma unroll
  for (int r = 0; r < 8; ++r) {
    const int M = r + hi * 8;
    dst[(r0 + M) * 64 + h0 + lm] = c[r] + bias;
  }
}

// ---------------------------------------------------------------------------
// K2: scores[q,k] = b2 + sum_h W2[h]*relu(qp[q,h]+kpb[k,h])
// One thread per k (kpb row in 64 VGPRs); qp row + W2 are wave-uniform
// (SGPR loads), so inner loop is 3 VALU ops per h. 32 q per block.
// ---------------------------------------------------------------------------
__global__ void scores_kernel(const float* __restrict__ qp,
                              const float* __restrict__ kpb,
                              const float* __restrict__ W2,
                              const float* __restrict__ b2,
                              float* __restrict__ scores) {
  const int k = blockIdx.x * 256 + threadIdx.x;   // [0,2048)
  const int q0 = blockIdx.y * 32;                 // [0,1024)
  const float b2v = b2[0];

  float kreg[64];
#pragma unroll
  for (int h = 0; h < 64; ++h) kreg[h] = kpb[k * 64 + h];

  for (int qi = 0; qi < 32; ++qi) {
    const int q = q0 + qi;
    const float* __restrict__ qrow = qp + q * 64;  // uniform -> s_load
    float acc = 0.0f;
#pragma unroll
    for (int h = 0; h < 64; ++h) {
      const float t = fmaxf(qrow[h] + kreg[h], 0.0f);
      acc = fmaf(W2[h], t, acc);                   // W2[h] uniform -> SGPR
    }
    scores[q * 2048 + k] = acc + b2v;
  }
}

// ---------------------------------------------------------------------------
// K3: per-row softmax stats (max, sum of exp) over 2048 cols. 1 block/row.
// ---------------------------------------------------------------------------
__global__ void stats_kernel(const float* __restrict__ scores,
                             float* __restrict__ rowmax,
                             float* __restrict__ rowsum) {
  __shared__ float red[256];
  const int q = blockIdx.x;
  const float* __restrict__ row = scores + q * 2048;

  float m = -INFINITY;
  for (int i = threadIdx.x; i < 2048; i += 256) m = fmaxf(m, row[i]);
  red[threadIdx.x] = m;
  __syncthreads();
  for (int s = 128; s > 0; s >>= 1) {
    if (threadIdx.x < s)
      red[threadIdx.x] = fmaxf(red[threadIdx.x], red[threadIdx.x + s]);
    __syncthreads();
  }
  m = red[0];
  __syncthreads();

  float sum = 0.0f;
  for (int i = threadIdx.x; i < 2048; i += 256) sum += __expf(row[i] - m);
  red[threadIdx.x] = sum;
  __syncthreads();
  for (int s = 128; s > 0; s >>= 1) {
    if (threadIdx.x < s) red[threadIdx.x] += red[threadIdx.x + s];
    __syncthreads();
  }
  if (threadIdx.x == 0) {
    rowmax[q] = m;
    rowsum[q] = red[0];
  }
}

// ---------------------------------------------------------------------------
// K4: partial[chunk][dd][n] = sum_{m in chunk} P[n,m]*value[dd,m]
// P[n,m] = exp(scores[n,m]-rowmax[n]) / rowsum[n], built on the fly into the
// WMMA A-fragment. One wave per (16-n tile, 256-m chunk); 4 accumulators
// cover all 64 dd columns (A reused across the 4 WMMAs).
// ---------------------------------------------------------------------------
__global__ void av_kernel(const float* __restrict__ scores,
                          const float* __restrict__ value,
                          const float* __restrict__ rowmax,
                          const float* __restrict__ rowsum,
                          float* __restrict__ partial) {
  const int n0 = blockIdx.x * 16;        // [0,1024)
  const int chunk = blockIdx.y;          // [0,8)
  const int m_base = chunk * 256;
  const int lane = threadIdx.x;
  const int lm = lane & 15;
  const int hi = lane >> 4;

  const int n = n0 + lm;                 // this lane's A-row (fixed)
  const float rmax = rowmax[n];
  const float rinv = 1.0f / rowsum[n];
  const float* __restrict__ srow = scores + n * 2048;

  v8f c0 = {}, c1 = {}, c2 = {}, c3 = {};
  for (int m0 = m_base; m0 < m_base + 256; m0 += 4) {
    const int ma = m0 + hi * 2;
    v2f a, bb0, bb1, bb2, bb3;
    a.x = __expf(srow[ma + 0] - rmax) * rinv;
    a.y = __expf(srow[ma + 1] - rmax) * rinv;
    bb0.x = value[(0  + lm) * 2048 + ma + 0];
    bb0.y = value[(0  + lm) * 2048 + ma + 1];
    bb1.x = value[(16 + lm) * 2048 + ma + 0];
    bb1.y = value[(16 + lm) * 2048 + ma + 1];
    bb2.x = value[(32 + lm) * 2048 + ma + 0];
    bb2.y = value[(32 + lm) * 2048 + ma + 1];
    bb3.x = value[(48 + lm) * 2048 + ma + 0];
    bb3.y = value[(48 + lm) * 2048 + ma + 1];
    c0 = wmma4(a, bb0, c0);
    c1 = wmma4(a, bb1, c1);
    c2 = wmma4(a, bb2, c2);
    c3 = wmma4(a, bb3, c3);
  }

  float* __restrict__ p = partial + chunk * 65536;
#pragma unroll
  for (int r = 0; r < 8; ++r) {
    const int M = r + hi * 8;            // n offset within tile
    p[(0  + lm) * 1024 + n0 + M] = c0[r];
    p[(16 + lm) * 1024 + n0 + M] = c1[r];
    p[(32 + lm) * 1024 + n0 + M] = c2[r];
    p[(48 + lm) * 1024 + n0 + M] = c3[r];
  }
}

// ---------------------------------------------------------------------------
// K5: out[dd*1024+n] = sum over 8 m-chunk partials (deterministic, no atomics)
// ---------------------------------------------------------------------------
__global__ void reduce_kernel(const float* __restrict__ partial,
                              float* __restrict__ out) {
  const int idx = blockIdx.x * 256 + threadIdx.x;  // [0,65536)
  float s = 0.0f;
#pragma unroll
  for (int c = 0; c < 8; ++c) s += partial[c * 65536 + idx];
  out[idx] = s;
}

// ---------------------------------------------------------------------------
extern "C" void kernel_launch(void* const* d_in, const int* in_sizes, int n_in,
                              void* d_out, int out_size, void* d_ws,
                              size_t ws_size, hipStream_t stream) {
  (void)in_sizes; (void)n_in; (void)out_size; (void)ws_size;
  const float* query = (const float*)d_in[0];   // (1,64,1024)
  const float* key   = (const float*)d_in[1];   // (1,64,2048)
  const float* value = (const float*)d_in[2];   // (1,64,2048)
  const float* W1    = (const float*)d_in[3];   // (64,128)
  const float* b1    = (const float*)d_in[4];   // (64,)
  const float* W2    = (const float*)d_in[5];   // (1,64)
  const float* b2    = (const float*)d_in[6];   // (1,)

  float* out    = (float*)d_out;        // 64*1024 floats
  float* scores = out + 64 * 1024;      // 1024*2048 floats

  float* ws      = (float*)d_ws;
  float* qp      = ws;                  //  65536 floats
  float* kpb     = ws + 65536;          // 131072 floats
  float* rowmax  = ws + 196608;         //   1024 floats
  float* rowsum  = rowmax + 1024;       //   1024 floats
  float* partial = rowsum + 1024;       // 8*65536 floats (~2 MB)

  proj_kernel<<<768, 32, 0, stream>>>(query, key, W1, b1, qp, kpb);
  scores_kernel<<<dim3(8, 32), 256, 0, stream>>>(qp, kpb, W2, b2, scores);
  stats_kernel<<<1024, 256, 0, stream>>>(scores, rowmax, rowsum);
  av_kernel<<<dim3(64, 8), 32, 0, stream>>>(scores, value, rowmax, rowsum,
                                            partial);
  reduce_kernel<<<256, 256, 0, stream>>>(partial, out);
}